// GATModel_45990509805610
// MI455X (gfx1250) — compile-verified
//
#include <hip/hip_runtime.h>
#include <hip/hip_bf16.h>

typedef float v2f __attribute__((ext_vector_type(2)));
typedef float v8f __attribute__((ext_vector_type(8)));

// ---------- helpers ----------
__device__ __forceinline__ unsigned fmap_ord(float f) {
  unsigned u = __float_as_uint(f);
  return (u & 0x80000000u) ? ~u : (u | 0x80000000u);
}
__device__ __forceinline__ float funmap_ord(unsigned u) {
  return (u & 0x80000000u) ? __uint_as_float(u & 0x7FFFFFFFu)
                           : __uint_as_float(~u);
}

// ---------- fills ----------
__global__ void fill_u32_kernel(unsigned* __restrict__ p, unsigned val, size_t count) {
  size_t i = (size_t)blockIdx.x * blockDim.x + threadIdx.x;
  size_t stride = (size_t)gridDim.x * blockDim.x;
  for (; i < count; i += stride) p[i] = val;
}

// ---------- fp32 WMMA GEMM: C[M,N] = A[M,K] * B[K,N] ----------
// One wave computes one 16x16 tile via V_WMMA_F32_16X16X4_F32 (exact fp32).
// Requires M%16==0, N%16==0, K%4==0, rows 8-byte aligned (K even).
__global__ void gemm_wmma_f32_kernel(const float* __restrict__ A,
                                     const float* __restrict__ B,
                                     float* __restrict__ C,
                                     int M, int N, int K) {
  const int lane   = threadIdx.x & 31;
  const int wave   = threadIdx.x >> 5;
  const int laneLo = lane & 15;
  const int laneHi = lane >> 4;           // 0 or 1
  const int mtile  = blockIdx.x;
  const int ntile  = blockIdx.y * (blockDim.x >> 5) + wave;
  if (ntile * 16 >= N) return;
  if (mtile * 16 >= M) return;

  const int row = mtile * 16 + laneLo;
  const int col = ntile * 16 + laneLo;
  // A fragment: lane holds A[row][k0 + 2*laneHi + {0,1}]
  const float* Arow = A + (size_t)row * K + 2 * laneHi;
  // B fragment: lane holds B[k0 + 2*laneHi + {0,1}][col]
  const float* Bcol = B + (size_t)(2 * laneHi) * N + col;

  v8f acc = {};
#pragma unroll 4
  for (int k0 = 0; k0 < K; k0 += 4) {
    v2f a = *(const v2f*)(Arow + k0);          // 8B aligned
    v2f b;
    b.x = Bcol[(size_t)k0 * N];
    b.y = Bcol[(size_t)(k0 + 1) * N];
    acc = __builtin_amdgcn_wmma_f32_16x16x4_f32(
        /*neg_a=*/false, a, /*neg_b=*/false, b,
        /*c_mod=*/(short)0, acc, /*reuse_a=*/false, /*reuse_b=*/false);
  }
  // C/D layout: VGPR r -> row = r + 8*laneHi, col = laneLo
  float* Crow = C + (size_t)(mtile * 16 + 8 * laneHi) * N + col;
#pragma unroll
  for (int r = 0; r < 8; ++r) Crow[(size_t)r * N] = acc[r];
}

// ---------- per-(node,head) attention score dots ----------
// 32-lane group per (node,head): a_src[n*H+h] = sum_c h[n][h][c]*att_src[h][c]
__global__ void att_scores_kernel(const float* __restrict__ h,
                                  const float* __restrict__ att_src,
                                  const float* __restrict__ att_dst,
                                  float* __restrict__ a_src,
                                  float* __restrict__ a_dst,
                                  int Nn, int H, int C) {
  int lane = threadIdx.x & 31;
  int grp  = (int)(((size_t)blockIdx.x * blockDim.x + threadIdx.x) >> 5);
  int total = Nn * H;
  if (grp >= total) return;
  int n = grp / H, hd = grp % H;
  const float* hp  = h + (size_t)n * H * C + (size_t)hd * C;
  const float* asp = att_src + (size_t)hd * C;
  const float* adp = att_dst + (size_t)hd * C;
  float ss = 0.f, sd = 0.f;
  for (int c = lane; c < C; c += 32) {
    float v = hp[c];
    ss += v * asp[c];
    sd += v * adp[c];
  }
#pragma unroll
  for (int off = 16; off > 0; off >>= 1) {
    ss += __shfl_down(ss, off, 32);
    sd += __shfl_down(sd, off, 32);
  }
  if (lane == 0) { a_src[grp] = ss; a_dst[grp] = sd; }
}

// ---------- edge phase: segment max of leaky-relu logits ----------
__global__ void edge_logit_max_kernel(const float* __restrict__ a_src,
                                      const float* __restrict__ a_dst,
                                      const int* __restrict__ src,
                                      const int* __restrict__ dst,
                                      unsigned* __restrict__ emax_u,
                                      int E, int H) {
  int tid = blockIdx.x * blockDim.x + threadIdx.x;
  if (tid >= E * H) return;
  int e = tid / H, hd = tid % H;
  int s = src[e], d = dst[e];
  float x = a_src[(size_t)s * H + hd] + a_dst[(size_t)d * H + hd];
  float logit = x > 0.f ? x : 0.2f * x;
  atomicMax(emax_u + (size_t)d * H + hd, fmap_ord(logit));
}

// unmap emax in place; non-finite (empty segment => -inf) -> 0 as in reference
__global__ void emax_fix_kernel(unsigned* __restrict__ buf, int count) {
  int i = blockIdx.x * blockDim.x + threadIdx.x;
  if (i >= count) return;
  float f = funmap_ord(buf[i]);
  unsigned bits = __float_as_uint(f);
  if ((bits & 0x7F800000u) == 0x7F800000u) f = 0.f;  // inf/nan -> 0
  ((float*)buf)[i] = f;
}

// ---------- edge phase: exp(logit - max) and segment sum ----------
__global__ void edge_exp_kernel(const float* __restrict__ a_src,
                                const float* __restrict__ a_dst,
                                const int* __restrict__ src,
                                const int* __restrict__ dst,
                                const float* __restrict__ emax_f,
                                float* __restrict__ ex,
                                float* __restrict__ denom,
                                int E, int H) {
  int tid = blockIdx.x * blockDim.x + threadIdx.x;
  if (tid >= E * H) return;
  int e = tid / H, hd = tid % H;
  int s = src[e], d = dst[e];
  float x = a_src[(size_t)s * H + hd] + a_dst[(size_t)d * H + hd];
  float logit = x > 0.f ? x : 0.2f * x;
  float v = expf(logit - emax_f[(size_t)d * H + hd]);
  ex[(size_t)e * H + hd] = v;
  unsafeAtomicAdd(denom + (size_t)d * H + hd, v);
}

// ---------- weighted gather/scatter aggregation (float4 vectorized) ----------
// Each thread handles 4 consecutive channels (one 128-bit gather + 4 atomics).
// threads per edge = HC/4; edges per 256-thread block = 1024/HC.
// Requires C % 4 == 0 (so 4 channels never straddle a head boundary).
__global__ void aggregate_kernel(const float* __restrict__ feat,
                                 const float* __restrict__ ex,
                                 const float* __restrict__ denom,
                                 const int* __restrict__ src,
                                 const int* __restrict__ dst,
                                 float* __restrict__ out,
                                 int E, int H, int C) {
  int HC   = H * C;
  int tpe  = HC >> 2;                      // threads per edge
  int epb  = blockDim.x / tpe;             // edges per block
  int e    = blockIdx.x * epb + (int)(threadIdx.x / tpe);
  int c4   = (threadIdx.x % tpe) << 2;     // starting channel (multiple of 4)
  if (e >= E) return;
  int s = src[e], d = dst[e];
  int hd = c4 / C;
  float alpha = ex[(size_t)e * H + hd] / (denom[(size_t)d * H + hd] + 1e-16f);
  const float4 f = *(const float4*)(feat + (size_t)s * HC + c4);  // 16B aligned
  float* op = out + (size_t)d * HC + c4;
  unsafeAtomicAdd(op + 0, f.x * alpha);
  unsafeAtomicAdd(op + 1, f.y * alpha);
  unsafeAtomicAdd(op + 2, f.z * alpha);
  unsafeAtomicAdd(op + 3, f.w * alpha);
}

// ---------- epilogues ----------
__global__ void elu_bias_kernel(float* __restrict__ h, const float* __restrict__ b,
                                size_t count, int HC) {
  size_t i = (size_t)blockIdx.x * blockDim.x + threadIdx.x;
  if (i >= count) return;
  float v = h[i] + b[i % HC];
  h[i] = v > 0.f ? v : expm1f(v);
}

__global__ void bias_add_kernel(float* __restrict__ o, const float* __restrict__ b,
                                size_t count, int C) {
  size_t i = (size_t)blockIdx.x * blockDim.x + threadIdx.x;
  if (i >= count) return;
  o[i] = o[i] + b[i % C];
}

// ---------- launch ----------
extern "C" void kernel_launch(void* const* d_in, const int* in_sizes, int n_in,
                              void* d_out, int out_size, void* d_ws, size_t ws_size,
                              hipStream_t stream) {
  const float* x        = (const float*)d_in[0];
  const float* W1       = (const float*)d_in[1];
  const float* att_src1 = (const float*)d_in[2];
  const float* att_dst1 = (const float*)d_in[3];
  const float* b1       = (const float*)d_in[4];
  const float* W2       = (const float*)d_in[5];
  const float* att_src2 = (const float*)d_in[6];
  const float* att_dst2 = (const float*)d_in[7];
  const float* b2       = (const float*)d_in[8];
  const int*   edge_idx = (const int*)d_in[9];

  const int IN  = 256;          // in_ch
  const int H1  = 4, C1 = 64;   // layer-1 heads/channels
  const int HC1 = H1 * C1;      // 256
  const int C2  = 32;           // layer-2 out channels (H=1)
  const int Nn  = in_sizes[0] / IN;       // 50000
  const int E   = in_sizes[9] / 2;        // 800000
  const int* src = edge_idx;
  const int* dst = edge_idx + E;

  float* out2 = (float*)d_out;                       // [Nn, 32]
  float* hout = (float*)d_out + (size_t)Nn * C2;     // [Nn, 256] (post-ELU h)

  // carve workspace
  char*  ws  = (char*)d_ws;
  size_t off = 0;
  auto carve = [&](size_t bytes) -> void* {
    void* p = ws + off;
    off += (bytes + 255) & ~(size_t)255;
    return p;
  };
  float*    h1      = (float*)   carve((size_t)Nn * HC1 * 4);
  float*    a_src1v = (float*)   carve((size_t)Nn * H1 * 4);
  float*    a_dst1v = (float*)   carve((size_t)Nn * H1 * 4);
  unsigned* emax1   = (unsigned*)carve((size_t)Nn * H1 * 4);
  float*    denom1  = (float*)   carve((size_t)Nn * H1 * 4);
  float*    ex1     = (float*)   carve((size_t)E * H1 * 4);
  float*    t2      = (float*)   carve((size_t)Nn * C2 * 4);
  float*    a_src2v = (float*)   carve((size_t)Nn * 4);
  float*    a_dst2v = (float*)   carve((size_t)Nn * 4);
  unsigned* emax2   = (unsigned*)carve((size_t)Nn * 4);
  float*    denom2  = (float*)   carve((size_t)Nn * 4);
  float*    ex2     = (float*)   carve((size_t)E * 4);

  const unsigned MAP_NEG_INF = 0x007FFFFFu;  // fmap_ord(-inf)

  // init: zero both d_out regions + denominators, set emax to map(-inf)
  fill_u32_kernel<<<4096, 256, 0, stream>>>((unsigned*)d_out, 0u, (size_t)Nn * (C2 + HC1));
  fill_u32_kernel<<<256, 256, 0, stream>>>((unsigned*)denom1, 0u, (size_t)Nn * H1);
  fill_u32_kernel<<<256, 256, 0, stream>>>((unsigned*)denom2, 0u, (size_t)Nn);
  fill_u32_kernel<<<256, 256, 0, stream>>>(emax1, MAP_NEG_INF, (size_t)Nn * H1);
  fill_u32_kernel<<<256, 256, 0, stream>>>(emax2, MAP_NEG_INF, (size_t)Nn);

  const int mtiles = Nn / 16;  // 3125 (Nn % 16 == 0 for this model)

  // ===== layer 1 =====
  // h1 = x @ W1
  {
    dim3 grid(mtiles, (HC1 / 16 + 7) / 8);
    gemm_wmma_f32_kernel<<<grid, 256, 0, stream>>>(x, W1, h1, Nn, HC1, IN);
  }
  // per-(node,head) attention dots
  {
    int groups = Nn * H1;
    int blocks = (int)(((size_t)groups * 32 + 255) / 256);
    att_scores_kernel<<<blocks, 256, 0, stream>>>(h1, att_src1, att_dst1,
                                                  a_src1v, a_dst1v, Nn, H1, C1);
  }
  // segment softmax
  {
    int blocks = (E * H1 + 255) / 256;
    edge_logit_max_kernel<<<blocks, 256, 0, stream>>>(a_src1v, a_dst1v, src, dst,
                                                      emax1, E, H1);
    emax_fix_kernel<<<(Nn * H1 + 255) / 256, 256, 0, stream>>>(emax1, Nn * H1);
    edge_exp_kernel<<<blocks, 256, 0, stream>>>(a_src1v, a_dst1v, src, dst,
                                                (const float*)emax1, ex1, denom1, E, H1);
  }
  // aggregation into hout: 64 threads/edge, 4 edges per block
  aggregate_kernel<<<(E + 3) / 4, 256, 0, stream>>>(h1, ex1, denom1, src, dst,
                                                    hout, E, H1, C1);
  // + b1, ELU (in place; hout is both output #2 and layer-2 input)
  {
    size_t count = (size_t)Nn * HC1;
    elu_bias_kernel<<<(int)((count + 255) / 256), 256, 0, stream>>>(hout, b1, count, HC1);
  }

  // ===== layer 2 =====
  // t2 = hout @ W2
  {
    dim3 grid(mtiles, 1);
    gemm_wmma_f32_kernel<<<grid, 64, 0, stream>>>(hout, W2, t2, Nn, C2, HC1);
  }
  {
    int blocks = (int)(((size_t)Nn * 32 + 255) / 256);
    att_scores_kernel<<<blocks, 256, 0, stream>>>(t2, att_src2, att_dst2,
                                                  a_src2v, a_dst2v, Nn, 1, C2);
  }
  {
    int blocks = (E + 255) / 256;
    edge_logit_max_kernel<<<blocks, 256, 0, stream>>>(a_src2v, a_dst2v, src, dst,
                                                      emax2, E, 1);
    emax_fix_kernel<<<(Nn + 255) / 256, 256, 0, stream>>>(emax2, Nn);
    edge_exp_kernel<<<blocks, 256, 0, stream>>>(a_src2v, a_dst2v, src, dst,
                                                (const float*)emax2, ex2, denom2, E, 1);
  }
  // aggregation into out2: 8 threads/edge, 32 edges per block
  aggregate_kernel<<<(E + 31) / 32, 256, 0, stream>>>(t2, ex2, denom2, src, dst,
                                                      out2, E, 1, C2);
  {
    size_t count = (size_t)Nn * C2;
    bias_add_kernel<<<(int)((count + 255) / 256), 256, 0, stream>>>(out2, b2, count, C2);
  }
}